// ROIAlign3D_13615046329083
// MI455X (gfx1250) — compile-verified
//
#include <hip/hip_runtime.h>

typedef __attribute__((ext_vector_type(2))) float v2f;
typedef __attribute__((ext_vector_type(8))) float v8f;

#define Bc    2
#define Cc    96
#define Dc    40
#define Hc    96
#define Wc    96
#define Kroi  256
#define HWc   (Hc * Wc)          /* 9216 */
#define DHWc  (Dc * HWc)         /* 368640 */
#define CDHWc (Cc * DHWc)        /* 35389440 */
#define NSLOT 28                 /* 14 x-samples * 2 corners */
#define HPIT  97                 /* LDS pitch to avoid bank conflicts */

// One block per (roi, pd, ph). 256 threads = 8 wave32.
// Stage A: compute sample coords / trilinear weights into LDS tables.
// Stage B: fold 4 z-corners x 4 y-corners (16 rows, weights wz*wy) into
//          H[s (28 x-slots)][c (96 ch)] via gathered f32 loads (mem-bound part).
// Stage C: out[pw, c] = sum_s A[pw,s] * H[s,c]  via V_WMMA_F32_16X16X4_F32,
//          A has wx/8 on rows pw (slots 4pw..4pw+3); 7 K=4 chunks cover s=0..27.
__global__ __launch_bounds__(256) void roialign3d_wmma_kernel(
    const float* __restrict__ in, const float* __restrict__ rois,
    float* __restrict__ out)
{
    __shared__ int   xs_lds[NSLOT];
    __shared__ float wxs_lds[NSLOT];
    __shared__ int   zidx_lds[4];
    __shared__ float wz_lds[4];
    __shared__ int   yidx_lds[4];
    __shared__ float wy_lds[4];
    __shared__ float wzy_lds[16];
    __shared__ int   rb_lds[16];
    __shared__ float Hs[NSLOT][HPIT];

    const int tid = threadIdx.x;
    const int k   = blockIdx.y;
    const int pd  = blockIdx.x / 7;
    const int ph  = blockIdx.x % 7;

    const float r0 = rois[k * 7 + 0];
    const float x1 = rois[k * 7 + 1] * 0.25f;
    const float y1 = rois[k * 7 + 2] * 0.25f;
    const float z1 = rois[k * 7 + 3] * 0.5f;
    const float x2 = rois[k * 7 + 4] * 0.25f;
    const float y2 = rois[k * 7 + 5] * 0.25f;
    const float z2 = rois[k * 7 + 6] * 0.5f;
    const int   bidx = (int)r0;
    const float rw = fmaxf(x2 - x1, 1.0f);
    const float rh = fmaxf(y2 - y1, 1.0f);
    const float rd = fmaxf(z2 - z1, 1.0f);

    // ---- Stage A: coordinate / weight tables ----
    if (tid < NSLOT) {                       // x slots: q = tid>>1, corner = tid&1
        int q = tid >> 1, corner = tid & 1;
        float coord = x1 + ((float)q + 0.5f) * 0.5f * (rw * (1.0f / 7.0f));
        bool  valid = (coord >= -1.0f) && (coord <= (float)Wc);
        float cc = fminf(fmaxf(coord, 0.0f), (float)(Wc - 1));
        int   lo = (int)floorf(cc);
        int   hi = min(lo + 1, Wc - 1);
        float fr = cc - (float)lo;
        xs_lds[tid]  = corner ? hi : lo;
        wxs_lds[tid] = valid ? (corner ? fr : 1.0f - fr) : 0.0f;
    } else if (tid >= 32 && tid < 36) {      // z pairs for qz = 2*pd + sz
        int t = tid - 32, sz = t >> 1, corner = t & 1;
        float coord = z1 + ((float)(2 * pd + sz) + 0.5f) * 0.5f * (rd * (1.0f / 7.0f));
        bool  valid = (coord >= -1.0f) && (coord <= (float)Dc);
        float cc = fminf(fmaxf(coord, 0.0f), (float)(Dc - 1));
        int   lo = (int)floorf(cc);
        int   hi = min(lo + 1, Dc - 1);
        float fr = cc - (float)lo;
        zidx_lds[t] = corner ? hi : lo;
        wz_lds[t]   = valid ? (corner ? fr : 1.0f - fr) : 0.0f;
    } else if (tid >= 36 && tid < 40) {      // y pairs for qy = 2*ph + sy
        int t = tid - 36, sy = t >> 1, corner = t & 1;
        float coord = y1 + ((float)(2 * ph + sy) + 0.5f) * 0.5f * (rh * (1.0f / 7.0f));
        bool  valid = (coord >= -1.0f) && (coord <= (float)Hc);
        float cc = fminf(fmaxf(coord, 0.0f), (float)(Hc - 1));
        int   lo = (int)floorf(cc);
        int   hi = min(lo + 1, Hc - 1);
        float fr = cc - (float)lo;
        yidx_lds[t] = corner ? hi : lo;
        wy_lds[t]   = valid ? (corner ? fr : 1.0f - fr) : 0.0f;
    }
    __syncthreads();

    if (tid < 16) {                          // combined z/y weights + row bases
        int zi = tid >> 2, yi = tid & 3;
        wzy_lds[tid] = wz_lds[zi] * wy_lds[yi];
        rb_lds[tid]  = bidx * CDHWc + zidx_lds[zi] * HWc + yidx_lds[yi] * Wc;
    }
    __syncthreads();

    // ---- Stage B: gather + z/y fold into H[s][c] ----
    float wj[16];
    int   rj[16];
#pragma unroll
    for (int j = 0; j < 16; ++j) { wj[j] = wzy_lds[j]; rj[j] = rb_lds[j]; }

    for (int e = tid; e < NSLOT * Cc; e += 256) {
        int s = e % NSLOT;                   // consecutive threads vary x-slot: locality
        int c = e / NSLOT;
        int xoff = xs_lds[s] + c * DHWc;
        float acc = 0.0f;
#pragma unroll
        for (int j = 0; j < 16; ++j) acc += wj[j] * in[rj[j] + xoff];
        Hs[s][c] = acc;
    }
    __syncthreads();

    // ---- Stage C: WMMA x-interp + pool. Waves 0..5 own 16-channel groups ----
    const int wave = tid >> 5;
    const int lane = tid & 31;
    if (wave < 6) {
        const int cbase = wave * 16;
        const int M     = lane & 15;         // A row (pw) / B column (channel)
        const int hh    = lane >> 4;         // half-wave: K offset {0,2}
        const int cIdx  = cbase + M;
        v8f acc = {};
#pragma unroll
        for (int kc = 0; kc < 7; ++kc) {
            const int s0 = 4 * kc + 2 * hh;  // K for VGPR0 of this chunk
            const int s1 = s0 + 1;           // K for VGPR1
            const float sel = (kc == M) ? 0.125f : 0.0f;  // A row pw==kc; /8 pooling
            v2f a, b;
            a[0] = wxs_lds[s0] * sel;
            a[1] = wxs_lds[s1] * sel;
            b[0] = Hs[s0][cIdx];
            b[1] = Hs[s1][cIdx];
            acc = __builtin_amdgcn_wmma_f32_16x16x4_f32(
                false, a, false, b, (short)0, acc, false, false);
        }
        if (lane < 16) {                     // D rows 0..6 live in VGPR 0..6, lanes 0..15
            float* op = out + (size_t)(k * Cc + cIdx) * 343 + pd * 49 + ph * 7;
#pragma unroll
            for (int v = 0; v < 7; ++v) op[v] = acc[v];
        }
    }
}

extern "C" void kernel_launch(void* const* d_in, const int* in_sizes, int n_in,
                              void* d_out, int out_size, void* d_ws, size_t ws_size,
                              hipStream_t stream) {
    (void)in_sizes; (void)n_in; (void)d_ws; (void)ws_size; (void)out_size;
    const float* in   = (const float*)d_in[0];
    const float* rois = (const float*)d_in[1];
    float*       out  = (float*)d_out;
    dim3 grid(7 * 7, Kroi, 1);   // blockIdx.x = pd*7+ph, blockIdx.y = roi
    dim3 block(256, 1, 1);
    roialign3d_wmma_kernel<<<grid, block, 0, stream>>>(in, rois, out);
}